// MambaDepthGating_51977694216649
// MI455X (gfx1250) — compile-verified
//
#include <hip/hip_runtime.h>
#include <hip/hip_bf16.h>
#include <math.h>

typedef float v2f __attribute__((ext_vector_type(2)));
typedef float v4f __attribute__((ext_vector_type(4)));
typedef float v8f __attribute__((ext_vector_type(8)));

#define D_MODEL 512
#define GATE 16
#define D_STATE 8
#define D_CONV 4
#define D_INNER 32
#define SEQ_L 8
#define BATCH_PER_WG 8
#define THREADS 128

__device__ __forceinline__ float sigmoidf_(float x) { return 1.0f / (1.0f + __expf(-x)); }
__device__ __forceinline__ float softplusf_(float x) { return (x > 20.0f) ? x : log1pf(__expf(x)); }

__global__ __launch_bounds__(THREADS)
void mamba_depth_gating_fused(
    const float* __restrict__ seq,     const float* __restrict__ deg,
    const float* __restrict__ W_down,  const float* __restrict__ W_deg,
    const float* __restrict__ b_deg,   const float* __restrict__ ln_g,
    const float* __restrict__ ln_b,    const float* __restrict__ W_logit,
    const float* __restrict__ b_logit, const float* __restrict__ W_in,
    const float* __restrict__ conv_w,  const float* __restrict__ conv_b,
    const float* __restrict__ W_xproj, const float* __restrict__ W_dt,
    const float* __restrict__ b_dt,    const float* __restrict__ A_log,
    const float* __restrict__ D_skip,  const float* __restrict__ W_out,
    float* __restrict__ out)
{
    __shared__ float g_lds[4][16][16];                     // biased gate activations, per wave tile
    __shared__ float xbuf[BATCH_PER_WG][SEQ_L][D_INNER];   // conv+silu output
    __shared__ float ybuf[BATCH_PER_WG][SEQ_L][D_INNER];   // scan output
    __shared__ float xdbl[BATCH_PER_WG][SEQ_L][20];        // dt_r | B | C (17, padded)
    __shared__ float lgbuf[BATCH_PER_WG][SEQ_L];
    __shared__ float wbuf[BATCH_PER_WG][SEQ_L];

    const int tid  = threadIdx.x;
    const int lane = tid & 31;
    const int wv   = tid >> 5;          // wave 0..3
    const int half = lane >> 4;         // 0/1
    const int l15  = lane & 15;

    // ---------------- Phase 1: g = seq @ W_down^T  (fp32 WMMA 16x16x4) ----------------
    // A (16x4 f32): lane m -> {A[m][k0], A[m][k0+1]}, lane m+16 -> {A[m][k0+2], A[m][k0+3]}
    // B (4x16 f32): lane n -> {B[k0][n], B[k0+1][n]} = {W_down[n][k0], W_down[n][k0+1]}, etc.
    const long rowBase = (long)blockIdx.x * 64 + wv * 16;   // flattened (b,l) row of first tile row
    const float* Ap = seq    + (rowBase + l15) * D_MODEL + half * 2;
    const float* Bp = W_down + (long)l15 * D_MODEL      + half * 2;

    v8f acc = {0.f, 0.f, 0.f, 0.f, 0.f, 0.f, 0.f, 0.f};
#pragma unroll 8
    for (int k = 0; k < D_MODEL; k += 4) {
        v2f a = *(const v2f*)(Ap + k);
        v2f b = *(const v2f*)(Bp + k);
        acc = __builtin_amdgcn_wmma_f32_16x16x4_f32(false, a, false, b, (short)0, acc, false, false);
    }

    // C/D layout: VGPR j, lanes 0-15 -> M=j ; lanes 16-31 -> M=j+8 ; N = lane&15
#pragma unroll
    for (int j = 0; j < 8; ++j) {
        const int  m  = j + half * 8;
        const long r  = rowBase + m;
        const long bg = r >> 3;                       // global batch of this row
        g_lds[wv][m][l15] = acc[j] + deg[bg] * W_deg[l15] + b_deg[l15];
    }
    __syncthreads();

    // ---------------- Phase 2: per-batch Mamba block ----------------
    const int bb   = tid >> 4;          // batch slot 0..7
    const int ln16 = tid & 15;
    const int gw   = bb >> 1;           // wave tile holding this batch
    const int go   = (bb & 1) * 8;      // row offset inside tile

    float xc[2][SEQ_L];                 // conv+silu (x)
    float zr[2][SEQ_L];                 // z branch
    {
        float xr[2][SEQ_L];
#pragma unroll
        for (int ci = 0; ci < 2; ++ci) {
            const int c = ln16 + ci * 16;
            const float* wiX = W_in + c * GATE;
            const float* wiZ = W_in + (c + D_INNER) * GATE;
#pragma unroll
            for (int l = 0; l < SEQ_L; ++l) {
                float sx = 0.f, sz = 0.f;
#pragma unroll
                for (int gi = 0; gi < GATE; ++gi) {
                    const float gv = g_lds[gw][go + l][gi];
                    sx += gv * wiX[gi];
                    sz += gv * wiZ[gi];
                }
                xr[ci][l] = sx; zr[ci][l] = sz;
            }
        }
        // causal depthwise conv + SiLU
#pragma unroll
        for (int ci = 0; ci < 2; ++ci) {
            const int c = ln16 + ci * 16;
#pragma unroll
            for (int l = 0; l < SEQ_L; ++l) {
                float s = conv_b[c];
#pragma unroll
                for (int k = 0; k < D_CONV; ++k) {
                    const int li = l + k - (D_CONV - 1);
                    if (li >= 0) s += conv_w[c * D_CONV + k] * xr[ci][li];
                }
                const float xs = s * sigmoidf_(s);
                xc[ci][l] = xs;
                xbuf[bb][l][c] = xs;
            }
        }
    }
    __syncthreads();

    // x_dbl = x @ W_xproj^T  (17 projections per l; lane p handles p, lane 0 also p=16)
    {
        const float* wp = W_xproj + ln16 * D_INNER;
#pragma unroll
        for (int l = 0; l < SEQ_L; ++l) {
            float s = 0.f;
#pragma unroll
            for (int d = 0; d < D_INNER; ++d) s += xbuf[bb][l][d] * wp[d];
            xdbl[bb][l][ln16] = s;
        }
        if (ln16 == 0) {
            const float* wp16 = W_xproj + 16 * D_INNER;
            for (int l = 0; l < SEQ_L; ++l) {
                float s = 0.f;
                for (int d = 0; d < D_INNER; ++d) s += xbuf[bb][l][d] * wp16[d];
                xdbl[bb][l][16] = s;
            }
        }
    }
    __syncthreads();

    // selective scan over L (2 channels per thread, state in registers)
    {
        float Ac[2][D_STATE], h[2][D_STATE], wdt[2], bdt[2], dsk[2];
#pragma unroll
        for (int ci = 0; ci < 2; ++ci) {
            const int c = ln16 + ci * 16;
            wdt[ci] = W_dt[c]; bdt[ci] = b_dt[c]; dsk[ci] = D_skip[c];
#pragma unroll
            for (int s = 0; s < D_STATE; ++s) { Ac[ci][s] = -__expf(A_log[c * D_STATE + s]); h[ci][s] = 0.f; }
        }
#pragma unroll
        for (int l = 0; l < SEQ_L; ++l) {
            const float dtr = xdbl[bb][l][0];
            float Bm[D_STATE], Cm[D_STATE];
#pragma unroll
            for (int s = 0; s < D_STATE; ++s) { Bm[s] = xdbl[bb][l][1 + s]; Cm[s] = xdbl[bb][l][1 + D_STATE + s]; }
#pragma unroll
            for (int ci = 0; ci < 2; ++ci) {
                const int c = ln16 + ci * 16;
                const float dt = softplusf_(dtr * wdt[ci] + bdt[ci]);
                const float xl = xc[ci][l];
                float y = 0.f;
#pragma unroll
                for (int s = 0; s < D_STATE; ++s) {
                    const float dA = __expf(dt * Ac[ci][s]);
                    h[ci][s] = dA * h[ci][s] + (dt * Bm[s]) * xl;
                    y += h[ci][s] * Cm[s];
                }
                y += dsk[ci] * xl;
                const float zl = zr[ci][l];
                y *= zl * sigmoidf_(zl);
                ybuf[bb][l][c] = y;
            }
        }
    }
    __syncthreads();

    // out-proj + residual + layernorm + logit (one l per lane, lanes 0..7)
    if (ln16 < SEQ_L) {
        const int l = ln16;
        float h2[GATE];
        float mu = 0.f;
#pragma unroll
        for (int gi = 0; gi < GATE; ++gi) {
            const float* wo = W_out + gi * D_INNER;
            float s = 0.f;
#pragma unroll
            for (int d = 0; d < D_INNER; ++d) s += ybuf[bb][l][d] * wo[d];
            h2[gi] = s + g_lds[gw][go + l][gi];
            mu += h2[gi];
        }
        mu *= (1.0f / GATE);
        float var = 0.f;
#pragma unroll
        for (int gi = 0; gi < GATE; ++gi) { const float dv = h2[gi] - mu; var += dv * dv; }
        var *= (1.0f / GATE);
        const float inv = rsqrtf(var + 1e-12f);
        float lg = b_logit[0];
#pragma unroll
        for (int gi = 0; gi < GATE; ++gi) {
            const float hn = (h2[gi] - mu) * inv * ln_g[gi] + ln_b[gi];
            lg += hn * W_logit[gi];
        }
        lgbuf[bb][l] = lg;
    }
    __syncthreads();

    // softmax over L (tiny; lane 0 of each batch group)
    if (ln16 == 0) {
        float m = lgbuf[bb][0];
        for (int l = 1; l < SEQ_L; ++l) m = fmaxf(m, lgbuf[bb][l]);
        float e[SEQ_L], s = 0.f;
        for (int l = 0; l < SEQ_L; ++l) { e[l] = __expf(lgbuf[bb][l] - m); s += e[l]; }
        const float rs = 1.0f / s;
        for (int l = 0; l < SEQ_L; ++l) wbuf[bb][l] = e[l] * rs;
    }
    __syncthreads();

    // ---------------- Phase 3: fused[b,:] = sum_l w[b,l] * seq[b,l,:] ----------------
    // All 128 threads sweep one batch row at a time: thread t owns floats [4t, 4t+4).
#pragma unroll
    for (int b2 = 0; b2 < BATCH_PER_WG; ++b2) {
        const long bg = (long)blockIdx.x * BATCH_PER_WG + b2;
        const v4f* srow = (const v4f*)(seq + bg * (SEQ_L * D_MODEL));
        v4f a = {0.f, 0.f, 0.f, 0.f};
#pragma unroll
        for (int l = 0; l < SEQ_L; ++l) a += wbuf[b2][l] * srow[l * (D_MODEL / 4) + tid];
        ((v4f*)(out + bg * D_MODEL))[tid] = a;
    }
}

extern "C" void kernel_launch(void* const* d_in, const int* in_sizes, int n_in,
                              void* d_out, int out_size, void* d_ws, size_t ws_size,
                              hipStream_t stream) {
    const float* seq     = (const float*)d_in[0];
    const float* deg     = (const float*)d_in[1];
    const float* W_down  = (const float*)d_in[2];
    const float* W_deg   = (const float*)d_in[3];
    const float* b_deg   = (const float*)d_in[4];
    const float* ln_g    = (const float*)d_in[5];
    const float* ln_b    = (const float*)d_in[6];
    const float* W_logit = (const float*)d_in[7];
    const float* b_logit = (const float*)d_in[8];
    const float* W_in    = (const float*)d_in[9];
    const float* conv_w  = (const float*)d_in[10];
    const float* conv_b  = (const float*)d_in[11];
    const float* W_xproj = (const float*)d_in[12];
    const float* W_dt    = (const float*)d_in[13];
    const float* b_dt    = (const float*)d_in[14];
    const float* A_log   = (const float*)d_in[15];
    const float* D_skip  = (const float*)d_in[16];
    const float* W_out   = (const float*)d_in[17];

    const int Btotal = in_sizes[1];                 // deg has B elements
    dim3 grid(Btotal / BATCH_PER_WG), block(THREADS);
    mamba_depth_gating_fused<<<grid, block, 0, stream>>>(
        seq, deg, W_down, W_deg, b_deg, ln_g, ln_b, W_logit, b_logit,
        W_in, conv_w, conv_b, W_xproj, W_dt, b_dt, A_log, D_skip, W_out,
        (float*)d_out);
}